// ImageResampleLayer_78838419685645
// MI455X (gfx1250) — compile-verified
//
#include <hip/hip_runtime.h>
#include <hip/hip_bf16.h>

#define AS1 __attribute__((address_space(1)))
#define AS3 __attribute__((address_space(3)))

typedef int v4i __attribute__((ext_vector_type(4)));

#if __has_builtin(__builtin_amdgcn_global_load_async_to_lds_b128)
#define HAVE_ASYNC_LDS 1
#else
#define HAVE_ASYNC_LDS 0
#endif

#if __has_builtin(__builtin_amdgcn_s_wait_asynccnt)
#define WAIT_ASYNC0() __builtin_amdgcn_s_wait_asynccnt(0)
#else
#define WAIT_ASYNC0() asm volatile("" ::: "memory")
#endif

// Problem dims (fixed by the reference)
constexpr int Bn = 2, H = 64, W = 64, D = 64, C = 32;
constexpr int WD = W * D;              // 4096
constexpr int HWD = H * W * D;         // 262144 samples per batch
constexpr int NSAMP = Bn * HWD;        // 524288 total samples
constexpr int SPB = 256;               // samples per block (never crosses batch)

__device__ __forceinline__ int iclamp(int v, int lo, int hi) {
    return v < lo ? lo : (v > hi ? hi : v);
}

__global__ __launch_bounds__(256)
void ImageResampleLayer_78838419685645_kernel(
    const float* __restrict__ feature,   // [B,H,W,D,C]
    const float* __restrict__ coord,     // [B,h,w,d,3]
    float* __restrict__ maskOut,         // [NSAMP]
    float* __restrict__ valOut)          // [NSAMP, C]
{
    __shared__ __align__(16) float sCoord[SPB * 3];   // 3 KB staged coords
    __shared__ __align__(16) float sW[SPB * 8];       // 8 weights / sample
    __shared__ __align__(16) int   sO[SPB * 8];       // 8 byte-offsets / sample

    const int tid = threadIdx.x;
    const int blk = blockIdx.x;

    // ---- Stage this block's coordinate tile (3072 B) into LDS via the
    // ---- CDNA5 async copy engine (GLOBAL_LOAD_ASYNC_TO_LDS_B128).
    {
        const char* src = (const char*)coord + (size_t)blk * (SPB * 3 * 4);
#if HAVE_ASYNC_LDS
        if (tid < (SPB * 3 * 4) / 16) {
            __builtin_amdgcn_global_load_async_to_lds_b128(
                (AS1 v4i*)(src + tid * 16),
                (AS3 v4i*)((char*)sCoord + tid * 16),
                /*imm offset=*/0, /*cpol=*/0);
        }
        WAIT_ASYNC0();
#else
        if (tid < (SPB * 3 * 4) / 16) {
            ((float4*)sCoord)[tid] = ((const float4*)src)[tid];
        }
#endif
    }
    __syncthreads();

    // ---- Phase 1: lane == sample. Weights, clamped corner offsets, mask.
    {
        const float cx = sCoord[tid * 3 + 0];
        const float cy = sCoord[tid * 3 + 1];
        const float cz = sCoord[tid * 3 + 2];

        const float fx = floorf(cx), fy = floorf(cy), fz = floorf(cz);
        const float tx = cx - fx, ty = cy - fy, tz = cz - fz;
        const int x0 = (int)fx, y0 = (int)fy, z0 = (int)fz;

        const float m = (cx >= 0.f && cx <= (float)(H - 1) &&
                         cy >= 0.f && cy <= (float)(W - 1) &&
                         cz >= 0.f && cz <= (float)(D - 1)) ? 1.f : 0.f;

        const int xc0 = iclamp(x0,     0, H - 1), xc1 = iclamp(x0 + 1, 0, H - 1);
        const int yc0 = iclamp(y0,     0, W - 1), yc1 = iclamp(y0 + 1, 0, W - 1);
        const int zc0 = iclamp(z0,     0, D - 1), zc1 = iclamp(z0 + 1, 0, D - 1);

        // Fold mask into the x-weights: value masking becomes free.
        const float wx0 = (1.f - tx) * m, wx1 = tx * m;
        const float wy0 = 1.f - ty,       wy1 = ty;
        const float wz0 = 1.f - tz,       wz1 = tz;

        const int bOff = (blk >> 10) * HWD;         // batch * 262144
        const int e00 = bOff + xc0 * WD + yc0 * D;
        const int e01 = bOff + xc0 * WD + yc1 * D;
        const int e10 = bOff + xc1 * WD + yc0 * D;
        const int e11 = bOff + xc1 * WD + yc1 * D;

        float* wp = &sW[tid * 8];
        int*   op = &sO[tid * 8];
        // byte offset = elementIndex * C * sizeof(float) = elem * 128
        wp[0] = wx0 * wy0 * wz0;  op[0] = (e00 + zc0) * 128;
        wp[1] = wx0 * wy0 * wz1;  op[1] = (e00 + zc1) * 128;
        wp[2] = wx0 * wy1 * wz0;  op[2] = (e01 + zc0) * 128;
        wp[3] = wx0 * wy1 * wz1;  op[3] = (e01 + zc1) * 128;
        wp[4] = wx1 * wy0 * wz0;  op[4] = (e10 + zc0) * 128;
        wp[5] = wx1 * wy0 * wz1;  op[5] = (e10 + zc1) * 128;
        wp[6] = wx1 * wy1 * wz0;  op[6] = (e11 + zc0) * 128;
        wp[7] = wx1 * wy1 * wz1;  op[7] = (e11 + zc1) * 128;

        __builtin_nontemporal_store(m, &maskOut[blk * SPB + tid]);
    }
    __syncthreads();

    // ---- Phase 2: lane == channel (32 lanes == C, wave32). Each of the 8
    // ---- waves in the block handles 32 samples; all per-sample scalars come
    // ---- from broadcast (conflict-free) LDS b128 loads.
    {
        const int lane = tid & 31;
        const int wv   = tid >> 5;
        const char* fB = (const char*)feature + (size_t)lane * 4;
        float* outBase = valOut + (size_t)(blk * SPB) * C + lane;

        #pragma unroll 4
        for (int i = 0; i < 32; ++i) {
            const int s = (wv << 5) | i;
            const float4 wa = *(const float4*)&sW[s * 8];
            const float4 wb = *(const float4*)&sW[s * 8 + 4];
            const int4   oa = *(const int4*)&sO[s * 8];
            const int4   ob = *(const int4*)&sO[s * 8 + 4];

            // 8 coalesced 128B gathers (one wave request per tap)
            const float f0 = *(const float*)(fB + oa.x);
            const float f1 = *(const float*)(fB + oa.y);
            const float f2 = *(const float*)(fB + oa.z);
            const float f3 = *(const float*)(fB + oa.w);
            const float f4 = *(const float*)(fB + ob.x);
            const float f5 = *(const float*)(fB + ob.y);
            const float f6 = *(const float*)(fB + ob.z);
            const float f7 = *(const float*)(fB + ob.w);

            float acc = wa.x * f0;
            acc = fmaf(wa.y, f1, acc);
            acc = fmaf(wa.z, f2, acc);
            acc = fmaf(wa.w, f3, acc);
            acc = fmaf(wb.x, f4, acc);
            acc = fmaf(wb.y, f5, acc);
            acc = fmaf(wb.z, f6, acc);
            acc = fmaf(wb.w, f7, acc);

            __builtin_nontemporal_store(acc, outBase + (size_t)s * C);
        }
    }
}

extern "C" void kernel_launch(void* const* d_in, const int* in_sizes, int n_in,
                              void* d_out, int out_size, void* d_ws, size_t ws_size,
                              hipStream_t stream) {
    (void)in_sizes; (void)n_in; (void)out_size; (void)d_ws; (void)ws_size;
    const float* feature = (const float*)d_in[0];
    const float* coord   = (const float*)d_in[1];
    float* maskOut = (float*)d_out;                 // first output: mask  [524288]
    float* valOut  = (float*)d_out + NSAMP;         // second output: value [524288*32]
    ImageResampleLayer_78838419685645_kernel<<<NSAMP / SPB, 256, 0, stream>>>(
        feature, coord, maskOut, valOut);
}